// MicroAdder_20280835572155
// MI455X (gfx1250) — compile-verified
//
#include <hip/hip_runtime.h>
#include <math.h>

typedef __attribute__((ext_vector_type(2))) float v2f;
typedef __attribute__((ext_vector_type(8))) float v8f;

#define T_SEQ 34
#define KDIM  48      // padded T for WMMA K/M
#define PITCH 50      // LDS row pitch in floats (even, avoids 64-bank conflicts)
#define VOCAB 10
#define WAVES 4

// ---- workspace table offsets (floats) ----
#define OFF_ATT   0                 // 48 rows x PITCH  (softmaxed causal attention, zero padded)
#define OFF_VTAB  (48*PITCH)        // 34*10*5 v lookup table  [s][digit][h]
#define OFF_TOK   (OFF_VTAB+1700)   // 10*2 token table
#define OFF_POS   (OFF_TOK+20)      // 34*3 positions
#define OFF_P     (OFF_POS+102)     // 5*10 combined final projection
#define OFF_A5    (OFF_P+50)        // out_proj_A (5)
#define OFF_B5    (OFF_A5+5)        // out_proj_B (5)
#define OFF_W5    (OFF_B5+5)        // norm_weight (5)
#define OFF_FC1   (OFF_W5+5)        // fc1_w (2x5)
#define OFF_WH    (OFF_FC1+10)      // head_proj_w (2x5)
#define TAB_TOT   (OFF_WH+10)       // 4307 floats

// flat position-table index per t: OFFS[SRC[t]] + IDX[t]
__constant__ int c_flat[T_SEQ] = {
  0,1,2,3,4,5,6,7,8,9, 11,
  0,1,2,3,4,5,6,7,8,9, 12,
  0,1,2,3,4,5,6,7,8,9, 10, 13 };

// -------------------- one-shot setup: all batch-invariant tables --------------------
__global__ void setup_tables(const float* tokA, const float* tokStart, const float* tokStride,
                             const float* zhi, const float* spe, const float* qw,
                             const float* qphase, const float* opA, const float* opB,
                             const float* nw, const float* fc1, const float* hw,
                             float* __restrict__ ws)
{
  __shared__ float s_pos[T_SEQ*3];
  __shared__ float s_invr[T_SEQ];
  __shared__ float s_q[T_SEQ*4];
  __shared__ float s_k[T_SEQ*4];
  __shared__ float s_tok[VOCAB*2];
  const int tid = threadIdx.x;
  const float A = tokA[0];

  if (tid < VOCAB) {
    float ang = tokStart[0] + (float)tid * tokStride[0];
    float cx = A*cosf(ang), sx = A*sinf(ang);
    s_tok[tid*2] = cx; s_tok[tid*2+1] = sx;
    ws[OFF_TOK+tid*2] = cx; ws[OFF_TOK+tid*2+1] = sx;
  }
  if (tid < T_SEQ) {
    int flat = c_flat[tid];
    float p0, p1, p2;
    if (flat < 10) {
      float a = 6.28318530717958647692f * (float)flat / 10.0f;
      p0 = 3.5f*cosf(a); p1 = 3.5f*sinf(a); p2 = 0.15f*(float)flat;
    } else if (flat == 10) { p0 = zhi[0]; p1 = zhi[1]; p2 = zhi[2]; }
    else if (flat == 12)   { p0 = spe[0]; p1 = spe[1]; p2 = spe[2]; }
    else { p0 = p1 = p2 = 0.f; }
    s_pos[tid*3]=p0; s_pos[tid*3+1]=p1; s_pos[tid*3+2]=p2;
    ws[OFF_POS+tid*3]=p0; ws[OFF_POS+tid*3+1]=p1; ws[OFF_POS+tid*3+2]=p2;
    // ||tok||^2 == A^2 for ANY digit -> rms is idx-independent
    float ss = A*A + p0*p0 + p1*p1 + p2*p2;
    s_invr[tid] = rsqrtf(ss*0.2f + 1e-5f);
  }
  __syncthreads();
  if (tid < T_SEQ) {
    float inv = s_invr[tid];
    float h0 = s_pos[tid*3+0]*inv*nw[2];
    float h1 = s_pos[tid*3+1]*inv*nw[3];
    float h2 = s_pos[tid*3+2]*inv*nw[4];
    float k0 = h0*qw[0] + h1*qw[1]  + h2*qw[2];
    float k1 = h0*qw[3] + h1*qw[4]  + h2*qw[5];
    float k2 = h0*qw[6] + h1*qw[7]  + h2*qw[8];
    float k3 = h0*qw[9] + h1*qw[10] + h2*qw[11];
    s_k[tid*4]=k0; s_k[tid*4+1]=k1; s_k[tid*4+2]=k2; s_k[tid*4+3]=k3;
    float c = cosf(qphase[0]), s = sinf(qphase[0]);   // k = pre-rotation q
    s_q[tid*4]   = k0*c - k1*s;
    s_q[tid*4+1] = k0*s + k1*c;
    s_q[tid*4+2] = k2*c - k3*s;
    s_q[tid*4+3] = k2*s + k3*c;
  }
  for (int i = tid; i < 48*PITCH; i += blockDim.x) ws[OFF_ATT+i] = 0.f;
  __syncthreads();
  if (tid < T_SEQ) {                       // one softmax row per thread (causal)
    const int t = tid;
    float l[T_SEQ];
    float mx = -1e30f;
    for (int s = 0; s <= t; ++s) {
      float e = 0.5f * (s_q[t*4]*s_k[s*4] + s_q[t*4+1]*s_k[s*4+1] +
                        s_q[t*4+2]*s_k[s*4+2] + s_q[t*4+3]*s_k[s*4+3]);
      l[s] = e; mx = fmaxf(mx, e);
    }
    float sum = 0.f;
    for (int s = 0; s <= t; ++s) { float p = expf(l[s]-mx); l[s] = p; sum += p; }
    float r = 1.0f / sum;
    for (int s = 0; s <= t; ++s) ws[OFF_ATT + t*PITCH + s] = l[s]*r;
  }
  // v lookup table: v_tab[s][d][h] = sum_c tok[d][c]*nw[c]*inv_rms[s]*hw[c][h]
  for (int e = tid; e < T_SEQ*VOCAB*5; e += blockDim.x) {
    int s = e/50, rr = e - s*50, d = rr/5, h = rr - d*5;
    ws[OFF_VTAB+e] = (s_tok[d*2]*nw[0]*hw[h] + s_tok[d*2+1]*nw[1]*hw[5+h]) * s_invr[s];
  }
  // P[i][d] = sum_c hw[c][i]*tok[d][c]  (head_proj_w.T @ tok_table.T collapsed)
  for (int e = tid; e < 50; e += blockDim.x) {
    int i = e/10, d = e - i*10;
    ws[OFF_P+e] = hw[i]*s_tok[d*2] + hw[5+i]*s_tok[d*2+1];
  }
  if (tid < 5)  { ws[OFF_A5+tid] = opA[tid]; ws[OFF_B5+tid] = opB[tid]; ws[OFF_W5+tid] = nw[tid]; }
  if (tid < 10) { ws[OFF_FC1+tid] = fc1[tid]; ws[OFF_WH+tid] = hw[tid]; }
}

// -------------------- main fused kernel: WMMA attention + epilogue --------------------
__global__ __launch_bounds__(128) void micro_tf(const int* __restrict__ idx,
                                                const float* __restrict__ ws,
                                                float* __restrict__ out, int B)
{
  __shared__ float tb[TAB_TOT];
  __shared__ float vT[WAVES][16*PITCH];   // per-wave B operand: [n][s], n = 3 seqs x 5 feats
  __shared__ float ol[WAVES][16*PITCH];   // per-wave D result:  [n][t]
  const int tid = threadIdx.x;
  for (int i = tid; i < TAB_TOT; i += blockDim.x) tb[i] = ws[i];
  const int wave = tid >> 5, lane = tid & 31;
  const int b0 = (blockIdx.x*WAVES + wave) * 3;
  float* vt  = vT[wave];
  float* olw = ol[wave];
  __syncthreads();

  // Build V stripe: column n = seqLocal*5 + h; rows s (zero pad s>=34, n==15, b>=B)
  for (int e = lane; e < 16*PITCH; e += 32) {
    int n = e / PITCH, s = e - n*PITCH;
    float v = 0.f;
    if (n < 15 && s < T_SEQ) {
      int b = b0 + n/5;
      if (b < B) {
        int d = idx[b*T_SEQ + s];
        v = tb[OFF_VTAB + s*50 + d*5 + (n - (n/5)*5)];
      }
    }
    vt[e] = v;
  }
  __syncthreads();

  // Fragment addressing (§7.12.2): lane L -> M/N = L&15; VGPR pair holds K {0,1} (L<16) or {2,3}
  const int nn = lane & 15;
  const int ks = (lane < 16) ? 0 : 2;

  v2f bf[12];                              // B frags: loaded once, reused for all 3 M-tiles
#pragma unroll
  for (int k = 0; k < 12; ++k)
    bf[k] = *(const v2f*)&vt[nn*PITCH + k*4 + ks];

#pragma unroll
  for (int m = 0; m < 3; ++m) {
    v8f acc = {};
    const float* arow = &tb[OFF_ATT + (m*16 + nn)*PITCH + ks];
#pragma unroll
    for (int k = 0; k < 12; ++k) {
      v2f a2 = *(const v2f*)&arow[k*4];    // ds_read_b64: att[t][k..k+1]
      acc = __builtin_amdgcn_wmma_f32_16x16x4_f32(
              false, a2, false, bf[k], (short)0, acc, false, false);
    }
    const int radd = m*16 + ((lane < 16) ? 0 : 8);  // D: VGPR j -> M = j / 8+j
#pragma unroll
    for (int j = 0; j < 8; ++j) olw[nn*PITCH + radd + j] = acc[j];
  }
  __syncthreads();

  // Epilogue: 3 seqs * 34 tokens = 102 tokens per wave
  for (int it = 0; it < 4; ++it) {
    int tk = it*32 + lane;
    if (tk >= 3*T_SEQ) break;
    int sl = tk / T_SEQ, t = tk - sl*T_SEQ;
    int b = b0 + sl;
    if (b >= B) continue;
    float o5[5], x[5];
#pragma unroll
    for (int h = 0; h < 5; ++h) o5[h] = olw[(sl*5 + h)*PITCH + t];
    int d = idx[b*T_SEQ + t];
    x[0] = tb[OFF_TOK + d*2];
    x[1] = tb[OFF_TOK + d*2 + 1];
#pragma unroll
    for (int j = 0; j < 3; ++j) x[2+j] = tb[OFF_POS + t*3 + j];
    // rank-1 out projection: x += (out . A) * B_row
    float s1 = 0.f;
#pragma unroll
    for (int h = 0; h < 5; ++h) s1 += o5[h]*tb[OFF_A5+h];
#pragma unroll
    for (int h = 0; h < 5; ++h) x[h] += s1*tb[OFF_B5+h];
    // ffn = gelu(rmsnorm(x) @ fc1.T) @ head_proj
    float ss = 0.f;
#pragma unroll
    for (int h = 0; h < 5; ++h) ss += x[h]*x[h];
    float inv = rsqrtf(ss*0.2f + 1e-5f);
    float f0 = 0.f, f1 = 0.f;
#pragma unroll
    for (int h = 0; h < 5; ++h) {
      float hv = x[h]*inv*tb[OFF_W5+h];
      f0 += hv*tb[OFF_FC1+h];
      f1 += hv*tb[OFF_FC1+5+h];
    }
    float g0 = 0.5f*f0*(1.0f + erff(f0*0.70710678118654752f));
    float g1 = 0.5f*f1*(1.0f + erff(f1*0.70710678118654752f));
#pragma unroll
    for (int h = 0; h < 5; ++h) x[h] += g0*tb[OFF_WH+h] + g1*tb[OFF_WH+5+h];
    // final rmsnorm then 5x10 combined projection
    ss = 0.f;
#pragma unroll
    for (int h = 0; h < 5; ++h) ss += x[h]*x[h];
    inv = rsqrtf(ss*0.2f + 1e-5f);
    float g5[5];
#pragma unroll
    for (int h = 0; h < 5; ++h) g5[h] = x[h]*inv*tb[OFF_W5+h];
    float* op = &out[(b*T_SEQ + t)*10];
#pragma unroll
    for (int dd = 0; dd < 10; ++dd) {
      float r = 0.f;
#pragma unroll
      for (int i = 0; i < 5; ++i) r += g5[i]*tb[OFF_P + i*10 + dd];
      op[dd] = r;
    }
  }
}

extern "C" void kernel_launch(void* const* d_in, const int* in_sizes, int n_in,
                              void* d_out, int out_size, void* d_ws, size_t ws_size,
                              hipStream_t stream) {
  const int*   idx       = (const int*)  d_in[0];
  const float* tokA      = (const float*)d_in[1];
  const float* tokStart  = (const float*)d_in[2];
  const float* tokStride = (const float*)d_in[3];
  const float* zhi       = (const float*)d_in[4];
  const float* spe       = (const float*)d_in[5];
  const float* qw        = (const float*)d_in[6];
  const float* qph       = (const float*)d_in[7];
  const float* opA       = (const float*)d_in[8];
  const float* opB       = (const float*)d_in[9];
  const float* nw        = (const float*)d_in[10];
  const float* fc1       = (const float*)d_in[11];
  const float* hw        = (const float*)d_in[12];
  float* ws  = (float*)d_ws;
  float* out = (float*)d_out;
  const int B = in_sizes[0] / T_SEQ;

  hipLaunchKernelGGL(setup_tables, dim3(1), dim3(128), 0, stream,
                     tokA, tokStart, tokStride, zhi, spe, qw, qph, opA, opB, nw, fc1, hw, ws);
  const int nTiles = (B + 2) / 3;                 // 3 sequences per wave (N packing)
  const int blocks = (nTiles + WAVES - 1) / WAVES;
  hipLaunchKernelGGL(micro_tf, dim3(blocks), dim3(128), 0, stream, idx, ws, out, B);
}